// CoGNN_7035156431215
// MI455X (gfx1250) — compile-verified
//
#include <hip/hip_runtime.h>
#include <hip/hip_bf16.h>
#include <stdint.h>

typedef float v2f __attribute__((ext_vector_type(2)));
typedef float v8f __attribute__((ext_vector_type(8)));
typedef unsigned u32x4 __attribute__((ext_vector_type(4)));
typedef int i32x4 __attribute__((ext_vector_type(4)));
typedef int i32x8 __attribute__((ext_vector_type(8)));

#define D_H   64
#define A_H   16
#define D_OUT 32

#if defined(__HIP_DEVICE_COMPILE__) && defined(__has_builtin)
#if __has_builtin(__builtin_amdgcn_tensor_load_to_lds) && __has_builtin(__builtin_amdgcn_s_wait_tensorcnt)
#define USE_TDM 1
#endif
#endif
#ifndef USE_TDM
#define USE_TDM 0
#endif

// ---------------------------------------------------------------------------
// Threefry-2x32 (JAX PRNG). 20 rounds, key injections every 4 rounds.
// ---------------------------------------------------------------------------
__host__ __device__ inline unsigned tf_rotl(unsigned v, int d) {
  return (v << d) | (v >> (32 - d));
}

__host__ __device__ inline void threefry2x32(unsigned k0, unsigned k1,
                                             unsigned x0, unsigned x1,
                                             unsigned* o0, unsigned* o1) {
  unsigned ks[3] = {k0, k1, k0 ^ k1 ^ 0x1BD11BDAu};
  unsigned v0 = x0 + ks[0], v1 = x1 + ks[1];
  const int ra[4] = {13, 15, 26, 6};
  const int rb[4] = {17, 29, 16, 24};
  for (int d = 1; d <= 5; ++d) {
    const int* R = (d & 1) ? ra : rb;
    for (int i = 0; i < 4; ++i) { v0 += v1; v1 = tf_rotl(v1, R[i]); v1 ^= v0; }
    v0 += ks[d % 3];
    v1 += ks[(d + 1) % 3] + (unsigned)d;
  }
  *o0 = v0; *o1 = v1;
}

// random_bits(key, 32, (n,2)) flattened index j in [0, 2n):
// counts split in halves; j<n -> o0 of block (j, j+n), else o1 of (j-n, j).
__device__ inline unsigned jax_bit_at(unsigned k0, unsigned k1, unsigned j, unsigned half) {
  unsigned o0, o1;
  if (j < half) { threefry2x32(k0, k1, j, j + half, &o0, &o1); return o0; }
  threefry2x32(k0, k1, j - half, j, &o0, &o1);
  return o1;
}

// uniform(1e-20, 1.0) -> gumbel noise, matching jax._src.random.uniform
__device__ inline float gumbel_from_bits(unsigned bits) {
  float f = __uint_as_float(0x3f800000u | (bits >> 9)) - 1.0f;
  float u = fmaxf(f * (1.0f - 1e-20f) + 1e-20f, 1e-20f);
  return -__logf(-__logf(u));
}

// ---------------------------------------------------------------------------
// Fp32 WMMA GEMM:  C[rows, F] = concat(A1[rows,K1], A2[rows,K2]) @ W + b
// Templated on (F, K1, K2) so the K loop fully unrolls and LDS addressing is
// immediate. Each wave owns a 16-row strip and ALL F/16 column tiles: one A
// fetch feeds TF = F/16 WMMAs (A reuse: global A traffic / TF).
// W + bias staged in LDS; the W panel is DMA'd by the Tensor Data Mover when
// available (tensor_load_to_lds + s_wait_tensorcnt), else cooperative copy.
// Operand layouts per CDNA5 ISA 7.12.2:
//   A (16x4 f32):  lanes 0-15 hold K={0,1}, lanes 16-31 hold K={2,3}
//   B (4x16 f32):  V0 = K rows 0/2, V1 = K rows 1/3 (lanes low/high)
//   C/D (16x16):   VGPR r -> M=r (lanes 0-15), M=r+8 (lanes 16-31)
// ---------------------------------------------------------------------------
template <int F, int K1, int K2, int RELU>
__global__ void gemm_wmma_f32(const float* __restrict__ A1,
                              const float* __restrict__ A2,
                              const float* __restrict__ W,
                              const float* __restrict__ bias,
                              float* __restrict__ C,
                              int tiles_n) {
  constexpr int K = K1 + K2;
  constexpr int TF = F / 16;
  __shared__ float smem[K * F + F];  // W panel then bias; LDS offset 0

#if USE_TDM
  // One wave issues the TDM descriptor: 2D tile F x K of 4-byte elements,
  // row stride F, destination LDS offset 0 (only shared object).
  if (threadIdx.x < 32) {
    unsigned long long ga = (unsigned long long)(uintptr_t)W;
    u32x4 g0 = {1u,                                     // count=1 valid D#
                0u,                                     // lds_addr = 0
                (unsigned)(ga & 0xFFFFFFFFu),           // global_addr[31:0]
                (unsigned)((ga >> 32) & 0x1FFFFFFu) | (2u << 30)};  // [56:32] | type=2
    i32x8 g1 = {(int)(2u << 16),                        // data_size = 4B
                (int)((unsigned)F << 16),               // tensor_dim0[15:0]
                (int)((unsigned)K << 16),               // tensor_dim1[15:0]
                (int)((unsigned)F << 16),               // tile_dim0 = F
                (int)(unsigned)K,                       // tile_dim1 = K
                (int)(unsigned)F,                       // tensor_dim0_stride = F
                0, 0};
    i32x4 gz = {0, 0, 0, 0};
#if __clang_major__ >= 23
    i32x8 gz8 = {0, 0, 0, 0, 0, 0, 0, 0};
    __builtin_amdgcn_tensor_load_to_lds(g0, g1, gz, gz, gz8, 0);
#else
    __builtin_amdgcn_tensor_load_to_lds(g0, g1, gz, gz, 0);
#endif
    __builtin_amdgcn_s_wait_tensorcnt(0);
  }
#else
  for (int i = threadIdx.x; i < K * F; i += blockDim.x) smem[i] = W[i];
#endif
  for (int i = threadIdx.x; i < F; i += blockDim.x) smem[K * F + i] = bias[i];
  __syncthreads();

  const int wave = threadIdx.x >> 5;
  const int strip = blockIdx.x * (blockDim.x >> 5) + wave;
  if (strip >= tiles_n) return;  // whole-wave uniform: EXEC all-1 for WMMA

  const int lane = threadIdx.x & 31;
  const int half = lane >> 4;  // 0: K pair {0,1}; 1: K pair {2,3}
  const int l    = lane & 15;
  const int row  = strip * 16 + l;

  v8f acc[TF];
#pragma unroll
  for (int cf = 0; cf < TF; ++cf) acc[cf] = (v8f){};

#pragma unroll
  for (int k = 0; k < K; k += 4) {
    const float* ap;
    if (K2 == 0 || k < K1) ap = A1 + (size_t)row * K1 + k;
    else                   ap = A2 + (size_t)row * K2 + (k - K1);
    const v2f a = *(const v2f*)(ap + 2 * half);
#pragma unroll
    for (int cf = 0; cf < TF; ++cf) {
      v2f b;
      b.x = smem[(k + 2 * half) * F + cf * 16 + l];
      b.y = smem[(k + 2 * half + 1) * F + cf * 16 + l];
      acc[cf] = __builtin_amdgcn_wmma_f32_16x16x4_f32(false, a, false, b,
                                                      (short)0, acc[cf],
                                                      false, false);
    }
  }

#pragma unroll
  for (int cf = 0; cf < TF; ++cf) {
    const int col = cf * 16 + l;
    const float bb = smem[K * F + col];
#pragma unroll
    for (int r = 0; r < 8; ++r) {
      int rr = strip * 16 + r + 8 * half;
      float v = acc[cf][r] + bb;
      if (RELU) v = fmaxf(v, 0.0f);
      C[(size_t)rr * F + col] = v;
    }
  }
}

// ---------------------------------------------------------------------------
// LayerNorm over 64 features, one wave32 per node (2 elems/lane).
// ---------------------------------------------------------------------------
__global__ void layernorm64_k(const float* __restrict__ in, float* __restrict__ out,
                              const float* __restrict__ g, const float* __restrict__ b,
                              int n) {
  int node = (int)((blockIdx.x * (size_t)blockDim.x + threadIdx.x) >> 5);
  int lane = threadIdx.x & 31;
  if (node >= n) return;
  const float* x = in + (size_t)node * D_H;
  float x0 = x[lane], x1 = x[lane + 32];
  float s = x0 + x1, sq = x0 * x0 + x1 * x1;
#pragma unroll
  for (int m = 16; m > 0; m >>= 1) {
    s  += __shfl_xor(s, m, 32);
    sq += __shfl_xor(sq, m, 32);
  }
  float mu = s * (1.0f / 64.0f);
  float var = sq * (1.0f / 64.0f) - mu * mu;
  float rs = rsqrtf(var + 1e-5f);
  float* o = out + (size_t)node * D_H;
  o[lane]      = (x0 - mu) * rs * g[lane] + b[lane];
  o[lane + 32] = (x1 - mu) * rs * g[lane + 32] + b[lane + 32];
}

// ---------------------------------------------------------------------------
// Unweighted 64-dim segment-sum: 16 lanes per edge, float4 gather + f32 atomics
// ---------------------------------------------------------------------------
__global__ void scatter64_k(const float* __restrict__ hn,
                            const int* __restrict__ src, const int* __restrict__ dst,
                            float* __restrict__ agg, int E) {
  unsigned tid = blockIdx.x * blockDim.x + threadIdx.x;
  if (tid >= (unsigned)E * 16u) return;
  int e = tid >> 4;
  int c = (tid & 15) << 2;
  int s = src[e], d = dst[e];
  float4 m = *(const float4*)(hn + (size_t)s * D_H + c);
  float* p = agg + (size_t)d * D_H + c;
  atomicAdd(p + 0, m.x); atomicAdd(p + 1, m.y);
  atomicAdd(p + 2, m.z); atomicAdd(p + 3, m.w);
}

// 16-dim segment-sum for both action hidden states at once: 4 lanes per edge.
__global__ void scatter16x2_k(const float* __restrict__ hin, const float* __restrict__ hout,
                              const int* __restrict__ src, const int* __restrict__ dst,
                              float* __restrict__ ain, float* __restrict__ aout, int E) {
  unsigned tid = blockIdx.x * blockDim.x + threadIdx.x;
  if (tid >= (unsigned)E * 4u) return;
  int e = tid >> 2;
  int c = (tid & 3) << 2;
  int s = src[e], d = dst[e];
  float4 mi = *(const float4*)(hin + (size_t)s * A_H + c);
  float4 mo = *(const float4*)(hout + (size_t)s * A_H + c);
  float* pi = ain + (size_t)d * A_H + c;
  float* po = aout + (size_t)d * A_H + c;
  atomicAdd(pi + 0, mi.x); atomicAdd(pi + 1, mi.y);
  atomicAdd(pi + 2, mi.z); atomicAdd(pi + 3, mi.w);
  atomicAdd(po + 0, mo.x); atomicAdd(po + 1, mo.y);
  atomicAdd(po + 2, mo.z); atomicAdd(po + 3, mo.w);
}

// ---------------------------------------------------------------------------
// Action-net conv2 (32->2) + Gumbel straight-through gate.
// Forward value of y + sg(y_hard - y) is exactly one_hot(argmax(logits + g));
// jnp.argmax ties -> index 0, so keep = (z0 >= z1).
// ---------------------------------------------------------------------------
__global__ void gate_k(const float* __restrict__ hin, const float* __restrict__ ain,
                       const float* __restrict__ hout, const float* __restrict__ aout,
                       const float* __restrict__ inW2, const float* __restrict__ inb2,
                       const float* __restrict__ outW2, const float* __restrict__ outb2,
                       float* __restrict__ in_keep, float* __restrict__ out_keep,
                       int n, unsigned ik0, unsigned ik1, unsigned ok0, unsigned ok1) {
  int i = blockIdx.x * blockDim.x + threadIdx.x;
  if (i >= n) return;
  float li0 = inb2[0], li1 = inb2[1];
  float lo0 = outb2[0], lo1 = outb2[1];
  const float* hi = hin  + (size_t)i * A_H;
  const float* ai = ain  + (size_t)i * A_H;
  const float* ho = hout + (size_t)i * A_H;
  const float* ao = aout + (size_t)i * A_H;
#pragma unroll
  for (int j = 0; j < A_H; ++j) {
    li0 += hi[j] * inW2[j * 2 + 0] + ai[j] * inW2[(A_H + j) * 2 + 0];
    li1 += hi[j] * inW2[j * 2 + 1] + ai[j] * inW2[(A_H + j) * 2 + 1];
    lo0 += ho[j] * outW2[j * 2 + 0] + ao[j] * outW2[(A_H + j) * 2 + 0];
    lo1 += ho[j] * outW2[j * 2 + 1] + ao[j] * outW2[(A_H + j) * 2 + 1];
  }
  unsigned half = (unsigned)n;
  float gi0 = gumbel_from_bits(jax_bit_at(ik0, ik1, 2u * (unsigned)i,      half));
  float gi1 = gumbel_from_bits(jax_bit_at(ik0, ik1, 2u * (unsigned)i + 1u, half));
  float go0 = gumbel_from_bits(jax_bit_at(ok0, ok1, 2u * (unsigned)i,      half));
  float go1 = gumbel_from_bits(jax_bit_at(ok0, ok1, 2u * (unsigned)i + 1u, half));
  in_keep[i]  = (li0 + gi0 >= li1 + gi1) ? 1.0f : 0.0f;
  out_keep[i] = (lo0 + go0 >= lo1 + go1) ? 1.0f : 0.0f;
}

// ---------------------------------------------------------------------------
// Weighted 64-dim segment-sum; also writes ews[e] = in_keep[dst]*out_keep[src].
// Skips atomics for dropped edges (w == 0), saving ~half the atomic traffic.
// ---------------------------------------------------------------------------
__global__ void wscatter64_k(const float* __restrict__ hn,
                             const int* __restrict__ src, const int* __restrict__ dst,
                             const float* __restrict__ in_keep,
                             const float* __restrict__ out_keep,
                             float* __restrict__ wagg, float* __restrict__ ews, int E) {
  unsigned tid = blockIdx.x * blockDim.x + threadIdx.x;
  if (tid >= (unsigned)E * 16u) return;
  int e = tid >> 4;
  int c = (tid & 15) << 2;
  int s = src[e], d = dst[e];
  float w = in_keep[d] * out_keep[s];
  if ((tid & 15) == 0) ews[e] = w;
  if (w != 0.0f) {
    float4 m = *(const float4*)(hn + (size_t)s * D_H + c);
    float* p = wagg + (size_t)d * D_H + c;
    atomicAdd(p + 0, m.x * w); atomicAdd(p + 1, m.y * w);
    atomicAdd(p + 2, m.z * w); atomicAdd(p + 3, m.w * w);
  }
}

// ---------------------------------------------------------------------------
extern "C" void kernel_launch(void* const* d_in, const int* in_sizes, int n_in,
                              void* d_out, int out_size, void* d_ws, size_t ws_size,
                              hipStream_t stream) {
  const float* x      = (const float*)d_in[0];
  const int*   ei     = (const int*)d_in[1];
  const float* W_enc  = (const float*)d_in[2];
  const float* b_enc  = (const float*)d_in[3];
  const float* env_W  = (const float*)d_in[4];
  const float* env_b  = (const float*)d_in[5];
  const float* W_dec  = (const float*)d_in[6];
  const float* b_dec  = (const float*)d_in[7];
  const float* ln_g   = (const float*)d_in[8];
  const float* ln_b   = (const float*)d_in[9];
  const float* in_W1  = (const float*)d_in[10];
  const float* in_b1  = (const float*)d_in[11];
  const float* in_W2  = (const float*)d_in[12];
  const float* in_b2  = (const float*)d_in[13];
  const float* out_W1 = (const float*)d_in[14];
  const float* out_b1 = (const float*)d_in[15];
  const float* out_W2 = (const float*)d_in[16];
  const float* out_b2 = (const float*)d_in[17];

  const int N = in_sizes[0] / D_H;   // 50000
  const int E = in_sizes[1] / 2;     // 800000
  const int* src = ei;
  const int* dst = ei + E;

  // Workspace carve-up (fp32)
  const size_t n64 = (size_t)N * D_H;
  const size_t n16 = (size_t)N * A_H;
  float* ws = (float*)d_ws;
  float* h        = ws; ws += n64;
  float* hn       = ws; ws += n64;
  float* agg64    = ws; ws += n64;
  float* wagg     = ws; ws += n64;
  float* hin      = ws; ws += n16;
  float* hout     = ws; ws += n16;
  float* ain      = ws; ws += n16;   // ain/aout contiguous (one memset)
  float* aout     = ws; ws += n16;
  float* in_keep  = ws; ws += N;
  float* out_keep = ws; ws += N;
  if ((size_t)((char*)ws - (char*)d_ws) > ws_size) return;

  float* out_mat = (float*)d_out;                 // [N, 32]
  float* ews_out = out_mat + (size_t)N * D_OUT;   // [3, E]

  const int TPB = 256;
  const int WPB = TPB / 32;
  const int tiles_n = N / 16;                     // 3125 (N divisible by 16)
  const int gemm_blocks = (tiles_n + WPB - 1) / WPB;

  // Encoder: h = x @ W_enc + b_enc
  gemm_wmma_f32<D_H, D_H, 0, 0><<<gemm_blocks, TPB, 0, stream>>>(
      x, nullptr, W_enc, b_enc, h, tiles_n);

  // Base threefry key for jax.random.key(1234) = (0, 1234)
  const unsigned bk0 = 0u, bk1 = 1234u;

  const int ln_blocks  = (int)(((size_t)N * 32 + TPB - 1) / TPB);
  const int s64_blocks = (int)(((size_t)E * 16 + TPB - 1) / TPB);
  const int s16_blocks = (int)(((size_t)E * 4 + TPB - 1) / TPB);
  const int nd_blocks  = (N + TPB - 1) / TPB;

  for (int l = 0; l < 3; ++l) {
    // h -> hn (layernorm)
    layernorm64_k<<<ln_blocks, TPB, 0, stream>>>(h, hn, ln_g, ln_b, N);

    // agg64 = segment_sum(hn[src] -> dst)
    hipMemsetAsync(agg64, 0, n64 * sizeof(float), stream);
    scatter64_k<<<s64_blocks, TPB, 0, stream>>>(hn, src, dst, agg64, E);

    // Action conv1 (shared agg64): hin/hout = relu(concat(hn, agg64) @ W1 + b1)
    gemm_wmma_f32<A_H, D_H, D_H, 1><<<gemm_blocks, TPB, 0, stream>>>(
        hn, agg64, in_W1, in_b1, hin, tiles_n);
    gemm_wmma_f32<A_H, D_H, D_H, 1><<<gemm_blocks, TPB, 0, stream>>>(
        hn, agg64, out_W1, out_b1, hout, tiles_n);

    // 16-dim segment sums of both hidden states
    hipMemsetAsync(ain, 0, 2 * n16 * sizeof(float), stream);
    scatter16x2_k<<<s16_blocks, TPB, 0, stream>>>(hin, hout, src, dst, ain, aout, E);

    // fold_in(key, 2l) / fold_in(key, 2l+1) on host (threefry of (0, data))
    unsigned ik0, ik1, ok0, ok1;
    threefry2x32(bk0, bk1, 0u, (unsigned)(2 * l),     &ik0, &ik1);
    threefry2x32(bk0, bk1, 0u, (unsigned)(2 * l + 1), &ok0, &ok1);

    gate_k<<<nd_blocks, TPB, 0, stream>>>(hin, ain, hout, aout,
                                          in_W2, in_b2, out_W2, out_b2,
                                          in_keep, out_keep, N, ik0, ik1, ok0, ok1);

    // Weighted aggregation + emit edge weights for this layer
    hipMemsetAsync(wagg, 0, n64 * sizeof(float), stream);
    wscatter64_k<<<s64_blocks, TPB, 0, stream>>>(hn, src, dst, in_keep, out_keep,
                                                 wagg, ews_out + (size_t)l * E, E);

    // Env conv: h = concat(hn, wagg) @ env_W[l] + env_b[l]
    gemm_wmma_f32<D_H, D_H, D_H, 0><<<gemm_blocks, TPB, 0, stream>>>(
        hn, wagg, env_W + (size_t)l * 2 * D_H * D_H, env_b + (size_t)l * D_H,
        h, tiles_n);
  }

  // Decoder: out = h @ W_dec + b_dec
  gemm_wmma_f32<D_OUT, D_H, 0, 0><<<gemm_blocks, TPB, 0, stream>>>(
      h, nullptr, W_dec, b_dec, out_mat, tiles_n);
}